// sparseGO_nn_29635274342875
// MI455X (gfx1250) — compile-verified
//
#include <hip/hip_runtime.h>
#include <math.h>

#define BATCH 256
#define GENE_DIM 3000
#define DRUG_DIM 2048
#define XDIM (GENE_DIM + DRUG_DIM)
#define OUT0 9000
#define OUT1 2400
#define OUT2 24
#define EPS 1e-5f

typedef float v2f __attribute__((ext_vector_type(2)));
typedef float v8f __attribute__((ext_vector_type(8)));

// ---------------------------------------------------------------------------
// Block-wide (256-thread) mean/var reduction helper. Each thread contributes
// one value; after return every thread has mu and var (biased, ddof=0).
// ---------------------------------------------------------------------------
__device__ inline void block_meanvar_256(float v, float& mu, float& var) {
    __shared__ float ssum[256];
    __shared__ float ssq[256];
    const int t = threadIdx.x;
    ssum[t] = v;
    ssq[t]  = v * v;
    __syncthreads();
    #pragma unroll
    for (int off = 128; off > 0; off >>= 1) {
        if (t < off) {
            ssum[t] += ssum[t + off];
            ssq[t]  += ssq[t + off];
        }
        __syncthreads();
    }
    mu  = ssum[0] * (1.0f / 256.0f);
    var = ssq[0] * (1.0f / 256.0f) - mu * mu;
}

// ---------------------------------------------------------------------------
// BatchNorm reading from the row-major input matrix x (B x XDIM), writing a
// column-major (feature, batch) output. grid.x = #features, block = 256.
// ---------------------------------------------------------------------------
__global__ __launch_bounds__(256) void bn_from_x_kernel(
    const float* __restrict__ x, int col_off,
    const float* __restrict__ g, const float* __restrict__ e,
    float* __restrict__ out)
{
    const int c = blockIdx.x;      // feature
    const int b = threadIdx.x;     // batch row
    const float v = x[(size_t)b * XDIM + col_off + c];
    float mu, var;
    block_meanvar_256(v, mu, var);
    out[(size_t)c * BATCH + b] = g[c] * (v - mu) * rsqrtf(var + EPS) + e[c];
}

// ---------------------------------------------------------------------------
// BatchNorm on a column-major (F x 256) buffer; optional tanh applied to the
// input value before statistics (for BN(tanh(splin(...)))). Safe in-place:
// every thread reads its element into a register before any write.
// grid.x = F, block = 256.
// ---------------------------------------------------------------------------
__global__ __launch_bounds__(256) void bn_cm_kernel(
    const float* __restrict__ in,
    const float* __restrict__ g, const float* __restrict__ e,
    float* __restrict__ out, int tanh_input)
{
    const int c = blockIdx.x;
    const int b = threadIdx.x;
    float v = in[(size_t)c * BATCH + b];
    if (tanh_input) v = tanhf(v);
    float mu, var;
    block_meanvar_256(v, mu, var);
    out[(size_t)c * BATCH + b] = g[c] * (v - mu) * rsqrtf(var + EPS) + e[c];
}

// ---------------------------------------------------------------------------
// Initialize splin output with broadcast bias. grid.x = n_out, block = 256.
// ---------------------------------------------------------------------------
__global__ __launch_bounds__(256) void fill_bias_kernel(
    float* __restrict__ out, const float* __restrict__ bias)
{
    out[(size_t)blockIdx.x * BATCH + threadIdx.x] = bias[blockIdx.x];
}

// ---------------------------------------------------------------------------
// Sparse linear scatter: one wave32 per edge; lanes sweep the 256-batch
// column with float4 (global_load_b128) gathers, then scatter with f32
// atomics (all L2-resident on MI455X's 192 MB L2).
// in/out are column-major (feature, 256). out[row, b] += in[col, b] * w.
// ---------------------------------------------------------------------------
__global__ __launch_bounds__(256) void splin_edge_kernel(
    const int* __restrict__ rows, const int* __restrict__ cols,
    const float* __restrict__ w,
    const float* __restrict__ in, float* __restrict__ out, int n_edges)
{
    const int wavesPerBlock = blockDim.x >> 5;
    const int edge = blockIdx.x * wavesPerBlock + (threadIdx.x >> 5);
    const int lane = threadIdx.x & 31;
    if (edge >= n_edges) return;
    const int   r  = rows[edge];
    const int   c  = cols[edge];
    const float wv = w[edge];
    const float4* __restrict__ src4 = (const float4*)(in + (size_t)c * BATCH);
    float*        __restrict__ dst  = out + (size_t)r * BATCH;
    #pragma unroll
    for (int q = lane; q < BATCH / 4; q += 32) {   // 2 iterations
        const float4 v = src4[q];
        const int b = q << 2;
        atomicAdd(&dst[b + 0], v.x * wv);
        atomicAdd(&dst[b + 1], v.y * wv);
        atomicAdd(&dst[b + 2], v.z * wv);
        atomicAdd(&dst[b + 3], v.w * wv);
    }
}

// ---------------------------------------------------------------------------
// Elementwise tanh copy (splin2 output -> concat rows).
// ---------------------------------------------------------------------------
__global__ __launch_bounds__(256) void tanh_copy_kernel(
    const float* __restrict__ in, float* __restrict__ out)
{
    const size_t i = (size_t)blockIdx.x * blockDim.x + threadIdx.x;
    out[i] = tanhf(in[i]);
}

// ---------------------------------------------------------------------------
// Dense GEMM with V_WMMA_F32_16X16X4_F32 (wave32, 16x16 C tile / wave),
// A tile staged through LDS and shared by all N-tile waves in the block.
//   act : column-major activations (K x 256), act[m,k] = act[k*256+m]
//   W   : row-major weights (N x K)
//   C[m,n] = sum_k act[m,k] * W[n,k] ; out[n,m] = tanh(C + bias[n]) (col-major)
// Launch: grid.x = 16 (M tiles of 256 batch rows), blockDim.x = 32 * (N/16);
// wave w owns N-tile w. Steady state: 64-wide K chunks (4 KB LDS A tile),
// fully-unrolled 16 WMMAs per barrier interval so loads/waits pipeline.
// Dynamic tail (K % 64, multiple of 4) only runs for the tiny K=56 GEMM.
// A fragment (16x4 f32, ISA layout): lanes 0-15 M=0..15 {v0=K0, v1=K1},
//                                    lanes 16-31       {v0=K2, v1=K3}.
// C layout: VGPR j, lanes 0-15 -> (M=j, N=lane); lanes 16-31 -> (M=8+j).
// ---------------------------------------------------------------------------
#define KCHUNK 64
__global__ __launch_bounds__(256) void wmma_gemm_lds_kernel(
    const float* __restrict__ act, const float* __restrict__ W,
    const float* __restrict__ bias, float* __restrict__ out,
    int N, int K, int apply_tanh)
{
    __shared__ float ldsA[16 * KCHUNK];   // layout [k][m], 16 floats per k-row

    const int tid   = threadIdx.x;
    const int lane  = tid & 31;
    const int wvid  = tid >> 5;            // wave id == N-tile index
    const int m0    = blockIdx.x << 4;     // M tile base (batch rows)
    const int n0    = wvid << 4;           // N tile base (features)
    const int l16   = lane & 15;
    const int khalf = (lane >> 4) << 1;    // 0 or 2

    const float* __restrict__ wBase = W + (size_t)(n0 + l16) * K;

    v8f c = {};
    int k0 = 0;
    // ---- steady state: full 64-wide chunks, compile-time inner loop -------
    for (; k0 + KCHUNK <= K; k0 += KCHUNK) {
        __syncthreads();                               // protect previous reads
        for (int i = tid; i < (KCHUNK << 4); i += blockDim.x) {
            const int k = i >> 4;
            const int m = i & 15;
            ldsA[(k << 4) + m] = act[(size_t)(k0 + k) * BATCH + m0 + m];
        }
        // Prefetch next chunk's strided activation rows while WMMAs run.
        if (k0 + KCHUNK < K && tid < KCHUNK && (k0 + KCHUNK + tid) < K) {
            __builtin_prefetch(act + (size_t)(k0 + KCHUNK + tid) * BATCH + m0, 0, 1);
        }
        __syncthreads();
        #pragma unroll
        for (int kk = 0; kk < KCHUNK; kk += 4) {
            v2f a, b;
            a.x = ldsA[((kk + khalf) << 4) + l16];
            a.y = ldsA[((kk + khalf + 1) << 4) + l16];
            b   = *(const v2f*)(wBase + k0 + kk + khalf);   // global_load_b64
            c = __builtin_amdgcn_wmma_f32_16x16x4_f32(
                    /*neg_a=*/false, a, /*neg_b=*/false, b,
                    /*c_mod=*/(short)0, c, /*reuse_a=*/false, /*reuse_b=*/false);
        }
    }
    // ---- tail: K % 64 (multiple of 4); only the K=56 head GEMM hits this --
    if (k0 < K) {
        const int kt = K - k0;
        __syncthreads();
        for (int i = tid; i < (kt << 4); i += blockDim.x) {
            const int k = i >> 4;
            const int m = i & 15;
            ldsA[(k << 4) + m] = act[(size_t)(k0 + k) * BATCH + m0 + m];
        }
        __syncthreads();
        for (int kk = 0; kk < kt; kk += 4) {
            v2f a, b;
            a.x = ldsA[((kk + khalf) << 4) + l16];
            a.y = ldsA[((kk + khalf + 1) << 4) + l16];
            b   = *(const v2f*)(wBase + k0 + kk + khalf);
            c = __builtin_amdgcn_wmma_f32_16x16x4_f32(
                    false, a, false, b, (short)0, c, false, false);
        }
    }

    const int   mbase = m0 + ((lane >> 4) << 3);
    const float bv    = bias[n0 + l16];
    float* __restrict__ o = out + (size_t)(n0 + l16) * BATCH + mbase;
    #pragma unroll
    for (int j = 0; j < 8; ++j) {
        float v = c[j] + bv;
        o[j] = apply_tanh ? tanhf(v) : v;
    }
}

// ---------------------------------------------------------------------------
// Head: o[b] = tanh(sum_n a[n,b]*Wa[n] + ba) * Wo + bo. grid=1, block=256.
// ---------------------------------------------------------------------------
__global__ __launch_bounds__(256) void final_head_kernel(
    const float* __restrict__ abn, const float* __restrict__ Wa,
    const float* __restrict__ ba, const float* __restrict__ Wo,
    const float* __restrict__ bo, float* __restrict__ out, int NF)
{
    const int b = threadIdx.x;
    float s = 0.0f;
    for (int n = 0; n < NF; ++n) {
        s += abn[(size_t)n * BATCH + b] * Wa[n];
    }
    const float o = tanhf(s + ba[0]);
    out[b] = o * Wo[0] + bo[0];
}

// ---------------------------------------------------------------------------
// Launcher
// ---------------------------------------------------------------------------
extern "C" void kernel_launch(void* const* d_in, const int* in_sizes, int n_in,
                              void* d_out, int out_size, void* d_ws, size_t ws_size,
                              hipStream_t stream) {
    const float* x     = (const float*)d_in[0];
    const int*   rows0 = (const int*)  d_in[1];
    const int*   cols0 = (const int*)  d_in[2];
    const float* w0    = (const float*)d_in[3];
    const float* b0    = (const float*)d_in[4];
    const int*   rows1 = (const int*)  d_in[5];
    const int*   cols1 = (const int*)  d_in[6];
    const float* w1    = (const float*)d_in[7];
    const float* b1    = (const float*)d_in[8];
    const int*   rows2 = (const int*)  d_in[9];
    const int*   cols2 = (const int*)  d_in[10];
    const float* w2    = (const float*)d_in[11];
    const float* b2    = (const float*)d_in[12];
    const float* gg    = (const float*)d_in[13];
    const float* eg    = (const float*)d_in[14];
    const float* g1    = (const float*)d_in[15];
    const float* e1    = (const float*)d_in[16];
    const float* g2    = (const float*)d_in[17];
    const float* e2    = (const float*)d_in[18];
    const float* Wd1   = (const float*)d_in[19];
    const float* bd1   = (const float*)d_in[20];
    const float* gd1   = (const float*)d_in[21];
    const float* ed1   = (const float*)d_in[22];
    const float* Wd2   = (const float*)d_in[23];
    const float* bd2   = (const float*)d_in[24];
    const float* gd2   = (const float*)d_in[25];
    const float* ed2   = (const float*)d_in[26];
    const float* Wd3   = (const float*)d_in[27];
    const float* bd3   = (const float*)d_in[28];
    const float* gd3   = (const float*)d_in[29];
    const float* ed3   = (const float*)d_in[30];
    const float* gf    = (const float*)d_in[31];
    const float* ef    = (const float*)d_in[32];
    const float* Wf    = (const float*)d_in[33];
    const float* bf    = (const float*)d_in[34];
    const float* ga    = (const float*)d_in[35];
    const float* ea    = (const float*)d_in[36];
    const float* Wa    = (const float*)d_in[37];
    const float* ba    = (const float*)d_in[38];
    const float* Wo    = (const float*)d_in[39];
    const float* bo    = (const float*)d_in[40];

    const int nE0 = in_sizes[1];
    const int nE1 = in_sizes[5];
    const int nE2 = in_sizes[9];

    float* ws = (float*)d_ws;
    // Persistent small region.
    float* concat = ws;                              // 56*256
    float* fa     = ws + 56 * BATCH;                 // 64*256
    float* big    = ws + (56 + 64) * BATCH;          // reusable big region

    // ---------------- Drug branch (runs first; big region reused later) ----
    float* dbn = big;                                // 2048*256
    float* d1  = big + (size_t)DRUG_DIM * BATCH;     // 128*256
    float* d2  = d1 + 128 * BATCH;                   // 64*256
    float* d3  = concat + OUT2 * BATCH;              // rows 24..55 of concat

    bn_from_x_kernel<<<DRUG_DIM, 256, 0, stream>>>(x, GENE_DIM, gd1, ed1, dbn);
    wmma_gemm_lds_kernel<<<16, 32 * (128 / 16), 0, stream>>>(dbn, Wd1, bd1, d1, 128, DRUG_DIM, 1);
    bn_cm_kernel<<<128, 256, 0, stream>>>(d1, gd2, ed2, d1, 0);
    wmma_gemm_lds_kernel<<<16, 32 * (64 / 16), 0, stream>>>(d1, Wd2, bd2, d2, 64, 128, 1);
    bn_cm_kernel<<<64, 256, 0, stream>>>(d2, gd3, ed3, d2, 0);
    wmma_gemm_lds_kernel<<<16, 32 * (32 / 16), 0, stream>>>(d2, Wd3, bd3, d3, 32, 64, 1);

    // ---------------- Gene branch (reuses big region) ----------------------
    float* hg = big;                                 // 3000*256
    float* s0 = big + (size_t)GENE_DIM * BATCH;      // 9000*256
    float* s1 = big;                                 // 2400*256 (hg retired)
    float* s2 = big + (size_t)GENE_DIM * BATCH;      // 24*256  (s0 retired)

    bn_from_x_kernel<<<GENE_DIM, 256, 0, stream>>>(x, 0, gg, eg, hg);

    fill_bias_kernel<<<OUT0, 256, 0, stream>>>(s0, b0);
    splin_edge_kernel<<<(nE0 + 7) / 8, 256, 0, stream>>>(rows0, cols0, w0, hg, s0, nE0);
    bn_cm_kernel<<<OUT0, 256, 0, stream>>>(s0, g1, e1, s0, 1);   // BN(tanh(.))

    fill_bias_kernel<<<OUT1, 256, 0, stream>>>(s1, b1);
    splin_edge_kernel<<<(nE1 + 7) / 8, 256, 0, stream>>>(rows1, cols1, w1, s0, s1, nE1);
    bn_cm_kernel<<<OUT1, 256, 0, stream>>>(s1, g2, e2, s1, 1);   // BN(tanh(.))

    fill_bias_kernel<<<OUT2, 256, 0, stream>>>(s2, b2);
    splin_edge_kernel<<<(nE2 + 7) / 8, 256, 0, stream>>>(rows2, cols2, w2, s1, s2, nE2);
    tanh_copy_kernel<<<OUT2, 256, 0, stream>>>(s2, concat);      // rows 0..23

    // ---------------- Head -------------------------------------------------
    bn_cm_kernel<<<56, 256, 0, stream>>>(concat, gf, ef, concat, 0);
    wmma_gemm_lds_kernel<<<16, 32 * (64 / 16), 0, stream>>>(concat, Wf, bf, fa, 64, 56, 1);
    bn_cm_kernel<<<64, 256, 0, stream>>>(fa, ga, ea, fa, 0);
    final_head_kernel<<<1, 256, 0, stream>>>(fa, Wa, ba, Wo, bo, (float*)d_out, 64);
}